// ASTE_SPAN_77721728188896
// MI455X (gfx1250) — compile-verified
//
#include <hip/hip_runtime.h>
#include <hip/hip_bf16.h>
#include <math.h>

// ---------------------------------------------------------------------------
// ASTE-span forward pass for MI455X (gfx1250, wave32, WMMA).
// Dead code in reference (att1 / attention_probs / states) is skipped.
// All matmuls use V_WMMA_F32_16X16X4_F32 (fp32 in/out -> matches reference
// precision). LSTM recurrences run as persistent single-workgroup kernels
// with h/c/gates resident in the 320KB LDS.
// ---------------------------------------------------------------------------

typedef __attribute__((ext_vector_type(2))) float v2f;
typedef __attribute__((ext_vector_type(8))) float v8f;

static constexpr int cVOCAB = 30000;
static constexpr int cEMB = 300;
static constexpr int cH = 300;     // hidden per direction
static constexpr int cG = 1200;    // 4*H gates
static constexpr int cD = 600;     // 2*H bidirectional
static constexpr int cLEN = 100;   // len_emb rows
static constexpr int cB = 32;
static constexpr int cS = 100;
static constexpr int cNS = 32;

__device__ __forceinline__ float sigf(float x) { return 1.0f / (1.0f + expf(-x)); }

__device__ __forceinline__ float wave_sum(float v) {
  #pragma unroll
  for (int off = 16; off; off >>= 1) v += __shfl_xor(v, off, 32);
  return v;
}

// ---------------------------------------------------------------------------
// Embedding gather: x[t, e] = emb[ids[t], e]
// ---------------------------------------------------------------------------
__global__ void embed_gather(const float* __restrict__ emb, const int* __restrict__ ids,
                             float* __restrict__ x, int n_tok, int E) {
  int i = blockIdx.x * blockDim.x + threadIdx.x;
  if (i >= n_tok * E) return;
  int t = i / E, e = i - t * E;
  x[i] = emb[(size_t)ids[t] * E + e];
}

// ---------------------------------------------------------------------------
// C[M,N] = A[M,K] @ W[N,K]^T + bias[N]   (fp32 WMMA 16x16x4, one wave / tile)
// M, N multiples of 16; K multiple of 4.
// A layout: lane l, vgpr j -> m = l&15, k = 2*(l>>4)+j
// B layout: lane l, vgpr j -> n = l&15, k = 2*(l>>4)+j  (B[k][n] = W[n][k])
// D layout: vgpr r, lane l -> m = r + 8*(l>>4), n = l&15
// ---------------------------------------------------------------------------
__global__ void gemm_bias_wmma(const float* __restrict__ A, const float* __restrict__ W,
                               const float* __restrict__ bias, float* __restrict__ C,
                               int M, int N, int K) {
  int wave = blockIdx.x * (blockDim.x >> 5) + (threadIdx.x >> 5);
  int lane = threadIdx.x & 31;
  int ntiles = N >> 4;
  int mtiles = M >> 4;
  if (wave >= mtiles * ntiles) return;       // wave-uniform exit, EXEC stays full
  int mt = wave / ntiles, nt = wave - mt * ntiles;
  int lr = lane & 15;
  int kh = (lane >> 4) << 1;
  const float* arow = A + (size_t)(mt * 16 + lr) * K + kh;
  const float* wrow = W + (size_t)(nt * 16 + lr) * K + kh;
  v8f acc = {};
  for (int k = 0; k < K; k += 4) {
    v2f a, b;
    a.x = arow[k]; a.y = arow[k + 1];
    b.x = wrow[k]; b.y = wrow[k + 1];
    acc = __builtin_amdgcn_wmma_f32_16x16x4_f32(false, a, false, b, (short)0, acc,
                                                false, false);
  }
  int n = nt * 16 + lr;
  float bv = bias ? bias[n] : 0.0f;
  int mbase = mt * 16 + ((lane >> 4) << 3);
  #pragma unroll
  for (int r = 0; r < 8; ++r)
    C[(size_t)(mbase + r) * N + n] = acc[r] + bv;
}

// ---------------------------------------------------------------------------
// Persistent LSTM recurrence, one 1024-thread workgroup per direction.
// blockIdx.x: 0 = forward, 1 = backward. pre: [B,S,G]; Whh: [G,H];
// hs: [B,S,2H] (forward cols 0..H-1, backward cols H..2H-1).
// h, c, and the full 32x1200 gate buffer live in LDS (~228 KB of the 320 KB).
// ---------------------------------------------------------------------------
__global__ __launch_bounds__(1024)
void lstm_recurrent(const float* __restrict__ pre_f, const float* __restrict__ pre_b,
                    const float* __restrict__ Whh_f, const float* __restrict__ Whh_b,
                    float* __restrict__ hs, int S) {
  __shared__ float h[cB][cH + 4];
  __shared__ float c[cB][cH + 4];
  __shared__ float g[cB][cG + 16];
  const int dir = blockIdx.x;
  const float* pre = dir ? pre_b : pre_f;
  const float* Whh = dir ? Whh_b : Whh_f;
  const int tid = threadIdx.x;
  const int lane = tid & 31;
  const int wave = tid >> 5;               // 0..31
  for (int i = tid; i < cB * cH; i += blockDim.x) {
    h[i / cH][i % cH] = 0.0f;
    c[i / cH][i % cH] = 0.0f;
  }
  __syncthreads();
  const int lr = lane & 15;
  const int kh = (lane >> 4) << 1;
  const int ntiles = cG / 16;              // 75
  for (int step = 0; step < S; ++step) {
    int t = dir ? (S - 1 - step) : step;
    // g = h @ Whh^T + pre[:, t, :]
    for (int tile = wave; tile < 2 * ntiles; tile += 32) {   // wave-uniform bound
      int mt = tile / ntiles;
      int nt = tile - mt * ntiles;
      const float* hrow = &h[mt * 16 + lr][kh];
      const float* wrow = Whh + (size_t)(nt * 16 + lr) * cH + kh;
      v8f acc = {};
      for (int k = 0; k < cH; k += 4) {
        v2f a, b;
        a.x = hrow[k]; a.y = hrow[k + 1];
        b.x = wrow[k]; b.y = wrow[k + 1];
        acc = __builtin_amdgcn_wmma_f32_16x16x4_f32(false, a, false, b, (short)0, acc,
                                                    false, false);
      }
      int n = nt * 16 + lr;
      int mbase = mt * 16 + ((lane >> 4) << 3);
      const float* prebase = pre + (size_t)t * cG + n;
      #pragma unroll
      for (int r = 0; r < 8; ++r) {
        int m = mbase + r;
        g[m][n] = acc[r] + prebase[(size_t)m * S * cG];
      }
    }
    __syncthreads();
    // gate nonlinearity + state update (PyTorch order i,f,g,o)
    for (int i = tid; i < cB * cH; i += blockDim.x) {
      int m = i / cH, j = i - m * cH;
      float gi = g[m][j];
      float gf = g[m][cH + j];
      float gg = g[m][2 * cH + j];
      float go = g[m][3 * cH + j];
      float cc = sigf(gf) * c[m][j] + sigf(gi) * tanhf(gg);
      float hh = sigf(go) * tanhf(cc);
      c[m][j] = cc;
      h[m][j] = hh;
      hs[((size_t)m * S + t) * cD + dir * cH + j] = hh;
    }
    __syncthreads();
  }
}

// ---------------------------------------------------------------------------
// prep: zero 8 accumulators, compute seq_len[b] = sum_s mask[b,s]
// accs: 0=loss_ote 1=loss_oe 2=loss_sp 3=p_loss 4=denom_sum 5=sq_sum 6=alpha 7=denom
// ---------------------------------------------------------------------------
__global__ void prep_kernel(const float* __restrict__ mask, float* __restrict__ seq_len,
                            float* __restrict__ accs) {
  int t = threadIdx.x;
  if (t < 8) accs[t] = 0.0f;
  if (t < cB) {
    float s = 0.0f;
    for (int j = 0; j < cS; ++j) s += mask[t * cS + j];
    seq_len[t] = s;
  }
}

// ---------------------------------------------------------------------------
// fc_ote / fc_oe cross-entropy, one wave per (b,s) row of ote_hidden
// ---------------------------------------------------------------------------
__global__ void fc_loss_kernel(const float* __restrict__ hid,
                               const float* __restrict__ Wote, const float* __restrict__ bote,
                               const float* __restrict__ Woe, const float* __restrict__ boe,
                               const int* __restrict__ ote, const int* __restrict__ oe,
                               const float* __restrict__ mask, const float* __restrict__ seq_len,
                               float* __restrict__ accs) {
  int row = blockIdx.x * (blockDim.x >> 5) + (threadIdx.x >> 5);
  int lane = threadIdx.x & 31;
  if (row >= cB * cS) return;
  int b = row / cS;
  const float* x = hid + (size_t)row * cD;
  float lg[7];
  #pragma unroll
  for (int cc = 0; cc < 5; ++cc) {
    const float* w = Wote + (size_t)cc * cD;
    float s = 0.0f;
    for (int k = lane; k < cD; k += 32) s += x[k] * w[k];
    lg[cc] = wave_sum(s) + bote[cc];
  }
  #pragma unroll
  for (int cc = 0; cc < 2; ++cc) {
    const float* w = Woe + (size_t)cc * cD;
    float s = 0.0f;
    for (int k = lane; k < cD; k += 32) s += x[k] * w[k];
    lg[5 + cc] = wave_sum(s) + boe[cc];
  }
  if (lane == 0) {
    float m5 = lg[0];
    for (int cc = 1; cc < 5; ++cc) m5 = fmaxf(m5, lg[cc]);
    float s5 = 0.0f;
    for (int cc = 0; cc < 5; ++cc) s5 += expf(lg[cc] - m5);
    float l_ote = (m5 + logf(s5)) - lg[ote[row]];
    float m2 = fmaxf(lg[5], lg[6]);
    float s2 = expf(lg[5] - m2) + expf(lg[6] - m2);
    float l_oe = (m2 + logf(s2)) - lg[5 + oe[row]];
    float scale = mask[row] / (seq_len[b] * (float)cB);
    atomicAdd(&accs[0], l_ote * scale);
    atomicAdd(&accs[1], l_oe * scale);
  }
}

// ---------------------------------------------------------------------------
// seq_states[b,n,d] = (seq_out[b,s0,d] + seq_out[b,s1,d]) / 2 ; spans_pos
// ---------------------------------------------------------------------------
__global__ void span_kernel(const float* __restrict__ seq_out, const int* __restrict__ spans,
                            float* __restrict__ seq_states, int* __restrict__ spans_pos) {
  int i = blockIdx.x * blockDim.x + threadIdx.x;
  if (i >= cB * cNS * cD) return;
  int d = i % cD;
  int bn = i / cD;
  int b = bn / cNS;
  int s0 = spans[bn * 2 + 0];
  int s1 = spans[bn * 2 + 1];
  seq_states[i] = 0.5f * (seq_out[((size_t)b * cS + s0) * cD + d] +
                          seq_out[((size_t)b * cS + s1) * cD + d]);
  if (d == 0) spans_pos[bn] = (int)(0.5f * (float)(s0 + s1));
}

// ---------------------------------------------------------------------------
// Small projection: out[r, c] = X[r,:] . W[c,:] + bias[c], one wave per row
// (used for SA = seq_states@att2^T+b, SB = seq_out@att2^T, SL = len_emb@att2^T)
// ---------------------------------------------------------------------------
__global__ void proj_small(const float* __restrict__ X, const float* __restrict__ W,
                           const float* __restrict__ bias, float* __restrict__ out,
                           int R, int C, int K) {
  int row = blockIdx.x * (blockDim.x >> 5) + (threadIdx.x >> 5);
  int lane = threadIdx.x & 31;
  if (row >= R) return;
  const float* x = X + (size_t)row * K;
  for (int cc = 0; cc < C; ++cc) {
    const float* w = W + (size_t)cc * K;
    float s = 0.0f;
    for (int k = lane; k < K; k += 32) s += x[k] * w[k];
    s = wave_sum(s);
    if (lane == 0) out[(size_t)row * C + cc] = s + (bias ? bias[cc] : 0.0f);
  }
}

// ---------------------------------------------------------------------------
// score: logits[b,n,s,c] = SA[b,n,c] + SB[b,s,c] + SL[|s-pos|,c]; softmax(5);
// s_w = sum(prob[1:]); loss_sp; running denom/sq sums. One thread per (b,n,s).
// ---------------------------------------------------------------------------
__global__ void score_kernel(const float* __restrict__ SA, const float* __restrict__ SB,
                             const float* __restrict__ SL, const int* __restrict__ spans_pos,
                             const int* __restrict__ oe_split, const float* __restrict__ mask,
                             const float* __restrict__ pol_mask, const float* __restrict__ seq_len,
                             float* __restrict__ s_w, float* __restrict__ accs) {
  int i = blockIdx.x * blockDim.x + threadIdx.x;
  int lane = threadIdx.x & 31;
  float sp = 0.0f, dn = 0.0f, sq = 0.0f;
  if (i < cB * cNS * cS) {
    int s = i % cS;
    int bn = i / cS;
    int b = bn / cNS;
    int dpos = s - spans_pos[bn];
    if (dpos < 0) dpos = -dpos;
    float lg[5];
    float mx = -1e30f;
    #pragma unroll
    for (int cc = 0; cc < 5; ++cc) {
      lg[cc] = SA[bn * 5 + cc] + SB[(b * cS + s) * 5 + cc] + SL[dpos * 5 + cc];
      mx = fmaxf(mx, lg[cc]);
    }
    float e[5], sum = 0.0f;
    #pragma unroll
    for (int cc = 0; cc < 5; ++cc) { e[cc] = expf(lg[cc] - mx); sum += e[cc]; }
    float inv = 1.0f / sum;
    float sw = (e[1] + e[2] + e[3] + e[4]) * inv;
    s_w[i] = sw;
    dn = sw;
    sq = sw * sw;
    int cls = oe_split[i];
    float logp = (lg[cls] - mx) - logf(sum);
    sp = -logp * mask[b * cS + s] * pol_mask[bn] / (seq_len[b] * (float)(cB * cNS));
  }
  sp = wave_sum(sp); dn = wave_sum(dn); sq = wave_sum(sq);
  if (lane == 0) {
    atomicAdd(&accs[2], sp);
    atomicAdd(&accs[4], dn);
    atomicAdd(&accs[5], sq);
  }
}

__global__ void alpha_kernel(float* __restrict__ accs) {
  float denom = accs[4] + 1.0f;
  accs[6] = accs[5] / denom;   // alpha
  accs[7] = denom;
}

// ---------------------------------------------------------------------------
// s2[b,n,d] = seq_states*(1-alpha) + (sum_s s_w[b,n,s]*seq_out[n,s,d]/denom)*alpha
// NOTE: reference einsum 'bns,nsd->bnd' indexes seq_out's batch dim by n.
// seq_out is L2-resident (7.7 MB), so the s-loop hits cache.
// ---------------------------------------------------------------------------
__global__ void s2_kernel(const float* __restrict__ seq_states, const float* __restrict__ seq_out,
                          const float* __restrict__ s_w, const float* __restrict__ accs,
                          float* __restrict__ s2) {
  int i = blockIdx.x * blockDim.x + threadIdx.x;
  if (i >= cB * cNS * cD) return;
  int d = i % cD;
  int bn = i / cD;
  int n = bn % cNS;
  float alpha = accs[6];
  float denom = accs[7];
  const float* sw = s_w + (size_t)bn * cS;
  const float* so = seq_out + (size_t)n * cS * cD + d;
  float acc = 0.0f;
  for (int s = 0; s < cS; ++s) acc += sw[s] * so[(size_t)s * cD];
  s2[i] = seq_states[i] * (1.0f - alpha) + (acc / denom) * alpha;
}

// ---------------------------------------------------------------------------
// p_loss: one wave per (b,n) row of pol_hidden; 3-class CE
// ---------------------------------------------------------------------------
__global__ void ploss_kernel(const float* __restrict__ hid, const float* __restrict__ pW,
                             const float* __restrict__ pb, const int* __restrict__ polarity,
                             const float* __restrict__ pol_mask, float* __restrict__ accs) {
  int row = blockIdx.x * (blockDim.x >> 5) + (threadIdx.x >> 5);
  int lane = threadIdx.x & 31;
  if (row >= cB * cNS) return;
  const float* x = hid + (size_t)row * cD;
  float lg[3];
  #pragma unroll
  for (int cc = 0; cc < 3; ++cc) {
    const float* w = pW + (size_t)cc * cD;
    float s = 0.0f;
    for (int k = lane; k < cD; k += 32) s += x[k] * w[k];
    lg[cc] = wave_sum(s) + pb[cc];
  }
  if (lane == 0) {
    float mx = fmaxf(lg[0], fmaxf(lg[1], lg[2]));
    float sum = expf(lg[0] - mx) + expf(lg[1] - mx) + expf(lg[2] - mx);
    float loss = (mx + logf(sum)) - lg[polarity[row]];
    atomicAdd(&accs[3], loss * pol_mask[row] / (float)(cB * cNS));
  }
}

__global__ void final_kernel(const float* __restrict__ accs, float* __restrict__ out) {
  out[0] = accs[0] + accs[1] + accs[2] + accs[3];
}

// ---------------------------------------------------------------------------
// Workspace layout (floats)
// ---------------------------------------------------------------------------
static constexpr size_t OFF_X     = 0;                          // 960000
static constexpr size_t OFF_PRE_F = OFF_X + 960000;             // 3840000
static constexpr size_t OFF_PRE_B = OFF_PRE_F + 3840000;        // 3840000
static constexpr size_t OFF_HID   = OFF_PRE_B + 3840000;        // 1920000 (ote_hidden)
static constexpr size_t OFF_SEQ   = OFF_HID + 1920000;          // 1920000 (seq_out)
static constexpr size_t OFF_SST   = OFF_SEQ + 1920000;          // 614400  (seq_states)
static constexpr size_t OFF_SW    = OFF_SST + 614400;           // 102400  (s_w)
static constexpr size_t OFF_S2    = OFF_SW + 102400;            // 614400
static constexpr size_t OFF_POLH  = OFF_S2 + 614400;            // 614400
static constexpr size_t OFF_SA    = OFF_POLH + 614400;          // 5120
static constexpr size_t OFF_SB    = OFF_SA + 5120;              // 16000
static constexpr size_t OFF_SL    = OFF_SB + 16000;             // 500
static constexpr size_t OFF_SEQL  = OFF_SL + 500;               // 32
static constexpr size_t OFF_ACCS  = OFF_SEQL + 32;              // 8
static constexpr size_t OFF_SPOS  = OFF_ACCS + 8;               // 1024 (ints)

extern "C" void kernel_launch(void* const* d_in, const int* in_sizes, int n_in,
                              void* d_out, int out_size, void* d_ws, size_t ws_size,
                              hipStream_t stream) {
  (void)in_sizes; (void)n_in; (void)out_size; (void)ws_size;
  // --- inputs, in setup_inputs() dict insertion order ---
  const float* word_embed = (const float*)d_in[0];
  const float* ote_f_Wih = (const float*)d_in[1];
  const float* ote_f_Whh = (const float*)d_in[2];
  const float* ote_f_b   = (const float*)d_in[3];
  const float* ote_b_Wih = (const float*)d_in[4];
  const float* ote_b_Whh = (const float*)d_in[5];
  const float* ote_b_b   = (const float*)d_in[6];
  const float* ctx_f_Wih = (const float*)d_in[7];
  const float* ctx_f_Whh = (const float*)d_in[8];
  const float* ctx_f_b   = (const float*)d_in[9];
  const float* ctx_b_Wih = (const float*)d_in[10];
  const float* ctx_b_Whh = (const float*)d_in[11];
  const float* ctx_b_b   = (const float*)d_in[12];
  const float* pol_f_Wih = (const float*)d_in[13];
  const float* pol_f_Whh = (const float*)d_in[14];
  const float* pol_f_b   = (const float*)d_in[15];
  const float* pol_b_Wih = (const float*)d_in[16];
  const float* pol_b_Whh = (const float*)d_in[17];
  const float* pol_b_b   = (const float*)d_in[18];
  const float* fc_ote_W  = (const float*)d_in[19];
  const float* fc_ote_b  = (const float*)d_in[20];
  const float* fc_oe_W   = (const float*)d_in[21];
  const float* fc_oe_b   = (const float*)d_in[22];
  const float* len_emb   = (const float*)d_in[23];
  // d_in[24] att1_W, d_in[25] att1_b: dead code in the reference loss
  const float* att2_W    = (const float*)d_in[26];
  const float* att2_b    = (const float*)d_in[27];
  const float* p_W       = (const float*)d_in[28];
  const float* p_b       = (const float*)d_in[29];
  const int*   inputs    = (const int*)d_in[30];
  // d_in[31] input_c: unused
  const float* attn_mask = (const float*)d_in[32];
  const int*   spans     = (const int*)d_in[33];
  const int*   polarity  = (const int*)d_in[34];
  const float* pol_mask  = (const float*)d_in[35];
  const int*   ote       = (const int*)d_in[36];
  const int*   oe        = (const int*)d_in[37];
  const int*   oe_split  = (const int*)d_in[38];

  float* ws = (float*)d_ws;
  float* x          = ws + OFF_X;
  float* pre_f      = ws + OFF_PRE_F;
  float* pre_b      = ws + OFF_PRE_B;
  float* ote_hidden = ws + OFF_HID;
  float* seq_out    = ws + OFF_SEQ;
  float* seq_states = ws + OFF_SST;
  float* s_w        = ws + OFF_SW;
  float* s2         = ws + OFF_S2;
  float* pol_hidden = ws + OFF_POLH;
  float* SA         = ws + OFF_SA;
  float* SB         = ws + OFF_SB;
  float* SL         = ws + OFF_SL;
  float* seq_len    = ws + OFF_SEQL;
  float* accs       = ws + OFF_ACCS;
  int*   spans_pos  = (int*)(ws + OFF_SPOS);

  auto gemm = [&](const float* A, const float* W, const float* bias, float* C,
                  int M, int N, int K) {
    int waves = (M / 16) * (N / 16);
    int blocks = (waves + 3) / 4;           // 4 waves (128 threads) per block
    gemm_bias_wmma<<<blocks, 128, 0, stream>>>(A, W, bias, C, M, N, K);
  };

  // 0) zero accumulators, seq_len
  prep_kernel<<<1, 32, 0, stream>>>(attn_mask, seq_len, accs);

  // 1) embedding gather: x [B,S,EMB]
  embed_gather<<<(cB * cS * cEMB + 255) / 256, 256, 0, stream>>>(
      word_embed, inputs, x, cB * cS, cEMB);

  // 2) ote BiLSTM
  gemm(x, ote_f_Wih, ote_f_b, pre_f, cB * cS, cG, cEMB);
  gemm(x, ote_b_Wih, ote_b_b, pre_b, cB * cS, cG, cEMB);
  lstm_recurrent<<<2, 1024, 0, stream>>>(pre_f, pre_b, ote_f_Whh, ote_b_Whh,
                                         ote_hidden, cS);

  // 3) fc_ote / fc_oe losses
  fc_loss_kernel<<<(cB * cS + 3) / 4, 128, 0, stream>>>(
      ote_hidden, fc_ote_W, fc_ote_b, fc_oe_W, fc_oe_b, ote, oe, attn_mask,
      seq_len, accs);

  // 4) ctx BiLSTM -> seq_out
  gemm(ote_hidden, ctx_f_Wih, ctx_f_b, pre_f, cB * cS, cG, cD);
  gemm(ote_hidden, ctx_b_Wih, ctx_b_b, pre_b, cB * cS, cG, cD);
  lstm_recurrent<<<2, 1024, 0, stream>>>(pre_f, pre_b, ctx_f_Whh, ctx_b_Whh,
                                         seq_out, cS);

  // 5) span pooling + attention scoring (qk never materialized)
  span_kernel<<<(cB * cNS * cD + 255) / 256, 256, 0, stream>>>(
      seq_out, spans, seq_states, spans_pos);
  proj_small<<<(cB * cNS + 3) / 4, 128, 0, stream>>>(seq_states, att2_W, att2_b,
                                                     SA, cB * cNS, 5, cD);
  proj_small<<<(cB * cS + 3) / 4, 128, 0, stream>>>(seq_out, att2_W, nullptr,
                                                    SB, cB * cS, 5, cD);
  proj_small<<<(cLEN + 3) / 4, 128, 0, stream>>>(len_emb, att2_W, nullptr,
                                                 SL, cLEN, 5, cD);
  score_kernel<<<(cB * cNS * cS + 255) / 256, 256, 0, stream>>>(
      SA, SB, SL, spans_pos, oe_split, attn_mask, pol_mask, seq_len, s_w, accs);
  alpha_kernel<<<1, 1, 0, stream>>>(accs);
  s2_kernel<<<(cB * cNS * cD + 255) / 256, 256, 0, stream>>>(
      seq_states, seq_out, s_w, accs, s2);

  // 6) pol BiLSTM over NS steps
  gemm(s2, pol_f_Wih, pol_f_b, pre_f, cB * cNS, cG, cD);
  gemm(s2, pol_b_Wih, pol_b_b, pre_b, cB * cNS, cG, cD);
  lstm_recurrent<<<2, 1024, 0, stream>>>(pre_f, pre_b, pol_f_Whh, pol_b_Whh,
                                         pol_hidden, cNS);

  // 7) polarity loss + final sum
  ploss_kernel<<<(cB * cNS + 3) / 4, 128, 0, stream>>>(
      pol_hidden, p_W, p_b, polarity, pol_mask, accs);
  final_kernel<<<1, 1, 0, stream>>>(accs, (float*)d_out);
}